// DifferentiableDDMTrainer_36112085025058
// MI455X (gfx1250) — compile-verified
//
#include <hip/hip_runtime.h>
#include <hip/hip_bf16.h>

// ---------------------------------------------------------------------------
// DDM first-passage simulation, memory-bound streaming kernel for gfx1250.
//   inputs : x[65536], noise[2000*65536], a, z, ndt, drift_gain (1 elem each)
//   output : d_out[0:65536) = pred_rt, d_out[65536:131072) = pred_choice
//
// Strategy: one thread per trial (2048 wave32 waves), 16-step-deep
// global->LDS async pipeline (ASYNCcnt), consuming 4 steps per ds_load_b128,
// wave-level early exit once all 32 trials of a wave are absorbed (cuts HBM
// traffic, the only thing that matters: 524 MB stream, ~22 us floor @23.3TB/s).
// ---------------------------------------------------------------------------

#define NTRIALS 65536
#define STEPS   2000
#define TPB     256
#define DEPTH   16          // async pipeline depth in steps
#define UNROLL  4           // steps consumed per ds_load_b128
#define GROUPS  (DEPTH / UNROLL)

#if defined(__has_builtin)
#  if __has_builtin(__builtin_amdgcn_global_load_async_to_lds_b32)
#    define USE_ASYNC 1
#  endif
#endif
#ifndef USE_ASYNC
#  define USE_ASYNC 0
#endif

// Per hipcc diagnostic: parameter 1 of the async builtin is a generic 'int *'.
#define AS1I(p) ((int*)(void*)(p))
#define AS3I(p) ((__attribute__((address_space(3))) int*)(p))

typedef float f4 __attribute__((ext_vector_type(4)));

__global__ __launch_bounds__(TPB) void ddm_scan_kernel(
    const float* __restrict__ x,
    const float* __restrict__ noise,
    const float* __restrict__ a_p,
    const float* __restrict__ z_p,
    const float* __restrict__ ndt_p,
    const float* __restrict__ gain_p,
    float* __restrict__ out)
{
    const int i = blockIdx.x * TPB + threadIdx.x;   // trial index

    const float a    = *a_p;
    const float z    = *z_p;
    const float ndt  = *ndt_p;
    const float gain = *gain_p;

    const float ns = sqrtf(0.001f);                 // SIGMA * sqrt(DT), folded
    const float dd = (gain * x[i]) * 0.001f;        // drift * DT (per step)
    const float up = a - 1e-6f;                     //  a - EPS
    const float lo = 1e-6f - a;                     // -a + EPS

    float dv  = z * a;
    float rt  = 0.0f;
    float ch  = 0.0f;
    bool  act = true;

#if USE_ASYNC
    // Each thread's 4 consecutive pipeline steps are contiguous 16B in LDS,
    // so 4 steps are consumed with a single ds_load_b128. Async b32 writes
    // land at 16B lane stride -> 32 distinct banks, conflict-free.
    __shared__ __align__(16) float buf[GROUPS][TPB][UNROLL];   // 16 KB
    const float* gcol = noise + i;                  // step t lives at gcol + t*NTRIALS

    #pragma unroll
    for (int p = 0; p < DEPTH; ++p) {
        __builtin_amdgcn_global_load_async_to_lds_b32(
            AS1I(gcol + (size_t)p * NTRIALS),
            AS3I(&buf[p / UNROLL][threadIdx.x][p % UNROLL]), 0, 0);
    }

    int  t    = 0;
    bool done = false;
    for (; t < STEPS - DEPTH; t += UNROLL) {
        // steps t..t+3 have landed when at most DEPTH-UNROLL remain in flight
        asm volatile("s_wait_asynccnt 12" ::: "memory");       // DEPTH-UNROLL
        const int g = (t >> 2) & (GROUPS - 1);
        f4 nz = *(const f4*)(&buf[g][threadIdx.x][0]);         // ds_load_b128
        // drain the ds read before overwriting this slot group:
        asm volatile("s_wait_dscnt 0" ::: "memory");
        #pragma unroll
        for (int u = 0; u < UNROLL; ++u) {
            __builtin_amdgcn_global_load_async_to_lds_b32(
                AS1I(gcol + (size_t)(t + DEPTH + u) * NTRIALS),
                AS3I(&buf[g][threadIdx.x][u]), 0, 0);
        }
        #pragma unroll
        for (int u = 0; u < UNROLL; ++u) {
            if (act) {
                dv = dv + dd + ns * nz[u];
                bool hu = dv >= up;
                bool hl = dv <= lo;
                if (hu || hl) {
                    rt  = (float)(t + u) * 0.001f + ndt;
                    ch  = hu ? 1.0f : 0.0f;
                    act = false;
                }
            }
        }
        if (!__any(act)) { done = true; break; }   // whole wave absorbed: stop streaming
    }

    if (!done) {
        asm volatile("s_wait_asynccnt 0" ::: "memory");        // drain tail
        for (; t < STEPS; t += UNROLL) {
            const int g = (t >> 2) & (GROUPS - 1);
            f4 nz = *(const f4*)(&buf[g][threadIdx.x][0]);
            #pragma unroll
            for (int u = 0; u < UNROLL; ++u) {
                if (act) {
                    dv = dv + dd + ns * nz[u];
                    bool hu = dv >= up;
                    bool hl = dv <= lo;
                    if (hu || hl) {
                        rt  = (float)(t + u) * 0.001f + ndt;
                        ch  = hu ? 1.0f : 0.0f;
                        act = false;
                    }
                }
            }
            if (!__any(act)) break;
        }
    }
#else
    // ---- fallback: plain streaming with non-temporal loads ----------------
    for (int t = 0; t < STEPS; ++t) {
        float nz = __builtin_nontemporal_load(noise + (size_t)t * NTRIALS + i);
        if (act) {
            dv = dv + dd + ns * nz;
            bool hu = dv >= up;
            bool hl = dv <= lo;
            if (hu || hl) {
                rt  = (float)t * 0.001f + ndt;
                ch  = hu ? 1.0f : 0.0f;
                act = false;
            }
        }
        if (((t & 15) == 15) && !__any(act)) break;
    }
#endif

    if (act) {                                      // never absorbed
        rt = 2.0f + ndt;                            // MAX_T + ndt
        ch = 0.5f;
    }
    out[i]           = rt;
    out[NTRIALS + i] = ch;
}

extern "C" void kernel_launch(void* const* d_in, const int* in_sizes, int n_in,
                              void* d_out, int out_size, void* d_ws, size_t ws_size,
                              hipStream_t stream) {
    const float* x     = (const float*)d_in[0];
    const float* noise = (const float*)d_in[1];
    const float* a     = (const float*)d_in[2];
    const float* z     = (const float*)d_in[3];
    const float* ndt   = (const float*)d_in[4];
    const float* gain  = (const float*)d_in[5];
    float* out = (float*)d_out;

    dim3 grid(NTRIALS / TPB);   // 256 blocks * 256 threads = 65536 = one thread/trial
    ddm_scan_kernel<<<grid, TPB, 0, stream>>>(x, noise, a, z, ndt, gain, out);
}